// MyGraphPooling_66039417143609
// MI455X (gfx1250) — compile-verified
//
#include <hip/hip_runtime.h>

// MI455X / gfx1250, wave32. Bandwidth-bound sort/segment-reduce pipeline.
// CDNA5-specific paths: TENSOR_LOAD_TO_LDS (TDM, TENSORcnt) tile staging,
// global_load_async_to_lds_b128 (ASYNCcnt) staging, s_wait_tensorcnt /
// s_wait_asynccnt, wave32 ballot radix ranking, global u32 atomic max carrying
// a monotonic f32 encoding, global_prefetch_b8.

typedef unsigned int u32;
typedef unsigned long long u64;
typedef long long i64;
typedef __attribute__((ext_vector_type(4))) u32 u32x4;
typedef __attribute__((ext_vector_type(8))) u32 u32x8;

#define TPB 256

static __device__ __forceinline__ void wait_async0() {
#if __has_builtin(__builtin_amdgcn_s_wait_asynccnt)
  __builtin_amdgcn_s_wait_asynccnt(0);
#else
  asm volatile("s_wait_asynccnt 0x0" ::: "memory");
#endif
}

static __device__ __forceinline__ void wait_tensor0() {
#if __has_builtin(__builtin_amdgcn_s_wait_tensorcnt)
  __builtin_amdgcn_s_wait_tensorcnt(0);
#else
  asm volatile("s_wait_tensorcnt 0x0" ::: "memory");
#endif
}

// ---------------------------------------------------------------- utilities
__global__ void fill_u32_kernel(u32* __restrict__ p, u32 v, u64 n) {
  u64 i = (u64)blockIdx.x * TPB + threadIdx.x;
  if (i < n) p[i] = v;
}

// per-dim max of quantized positions (pool = 4 -> >>2, pos >= 0)
__global__ void qmax_kernel(const i64* __restrict__ pos, u32* __restrict__ dmax, u32 N) {
  __shared__ u32 sm[3];
  if (threadIdx.x < 3) sm[threadIdx.x] = 0;
  __syncthreads();
  u32 i = blockIdx.x * TPB + threadIdx.x;
  if (i < N) {
    atomicMax(&sm[0], (u32)(pos[3ull * i + 0] >> 2));
    atomicMax(&sm[1], (u32)(pos[3ull * i + 1] >> 2));
    atomicMax(&sm[2], (u32)(pos[3ull * i + 2] >> 2));
  }
  __syncthreads();
  if (threadIdx.x < 3) atomicMax(&dmax[threadIdx.x], sm[threadIdx.x]);
}

__global__ void make_keys_kernel(const i64* __restrict__ pos, const i64* __restrict__ batch,
                                 const u32* __restrict__ dmax, u32* __restrict__ keys,
                                 u32* __restrict__ idx, u32 N, u32 Npad) {
  u32 i = blockIdx.x * TPB + threadIdx.x;
  if (i >= Npad) return;
  if (i < N) {
    u32 kx = dmax[0] + 1u, ky = dmax[1] + 1u, kz = dmax[2] + 1u;
    u32 q0 = (u32)(pos[3ull * i + 0] >> 2);
    u32 q1 = (u32)(pos[3ull * i + 1] >> 2);
    u32 q2 = (u32)(pos[3ull * i + 2] >> 2);
    u32 b  = (u32)batch[i];
    keys[i] = ((b * kx + q0) * ky + q1) * kz + q2;   // fits 32 bits (<=2^25 here)
    idx[i]  = i;
  } else {
    keys[i] = 0xFFFFFFFFu;  // pad sorts last
    idx[i]  = 0u;
  }
}

// ------------------------------------------------------------- radix sort
template <typename K>
__global__ void radix_hist_kernel(const K* __restrict__ keys, u32* __restrict__ hist,
                                  u32 numBlocks, u32 shift) {
  __shared__ u32 cnt[256];
  cnt[threadIdx.x] = 0;
  __syncthreads();
  u32 i = blockIdx.x * TPB + threadIdx.x;
  // stream-ahead prefetch (clamped; lowers to global_prefetch_b8)
  u32 pf = i + 64u * TPB;
  u32 lim = numBlocks * 256u;
  if (pf >= lim) pf = lim - 1u;
  __builtin_prefetch(&keys[pf], 0, 0);
  K k = keys[i];  // padded input: always in range
  u32 d = (u32)((k >> shift) & (K)0xFF);
  atomicAdd(&cnt[d], 1u);
  __syncthreads();
  hist[(u64)threadIdx.x * numBlocks + blockIdx.x] = cnt[threadIdx.x];
}

// stable scatter: wave32 ballot peer-matching + cross-wave LDS offsets
template <typename K, bool P>
__global__ void radix_scatter_kernel(const K* __restrict__ keys_in, const u32* __restrict__ pay_in,
                                     K* __restrict__ keys_out, u32* __restrict__ pay_out,
                                     const u32* __restrict__ hist_scan, u32 numBlocks, u32 shift) {
  __shared__ u32 wdc[8 * 256];  // per-wave per-digit counts
  for (u32 k = threadIdx.x; k < 2048u; k += TPB) wdc[k] = 0;
  __syncthreads();
  u32 i = blockIdx.x * TPB + threadIdx.x;
  K key = keys_in[i];
  u32 pay = P ? pay_in[i] : 0u;
  u32 d = (u32)((key >> shift) & (K)0xFF);
  u32 lane = threadIdx.x & 31u;
  u32 wave = threadIdx.x >> 5;
  u32 peers = 0xFFFFFFFFu;
#pragma unroll
  for (int b = 0; b < 8; ++b) {
    u32 bit = (d >> b) & 1u;
    u32 vote = (u32)__ballot(bit != 0u);  // wave32: low 32 bits
    peers &= bit ? vote : ~vote;
  }
  u32 ltmask = (lane == 0u) ? 0u : (0xFFFFFFFFu >> (32u - lane));
  u32 rank = __popc(peers & ltmask);
  if (rank == 0u) wdc[wave * 256u + d] = (u32)__popc(peers);  // group leader
  __syncthreads();
  u32 before = 0;
  for (u32 w = 0; w < wave; ++w) before += wdc[w * 256u + d];
  u32 dst = hist_scan[(u64)d * numBlocks + blockIdx.x] + before + rank;
  keys_out[dst] = key;
  if (P) pay_out[dst] = pay;
}

// ------------------------------------------------------- exclusive scan (u32)
__global__ void scan_blocks_kernel(const u32* __restrict__ in, u32* __restrict__ out,
                                   u32* __restrict__ partials, u32 n) {
  __shared__ u32 sh[TPB];
  u32 base = blockIdx.x * 1024u;
  u32 v[4]; u32 tsum = 0;
#pragma unroll
  for (int k = 0; k < 4; ++k) {
    u32 i = base + threadIdx.x * 4u + k;
    v[k] = (i < n) ? in[i] : 0u;
    tsum += v[k];
  }
  sh[threadIdx.x] = tsum;
  __syncthreads();
  for (u32 o = 1; o < TPB; o <<= 1) {
    u32 t = (threadIdx.x >= o) ? sh[threadIdx.x - o] : 0u;
    __syncthreads();
    sh[threadIdx.x] += t;
    __syncthreads();
  }
  u32 run = sh[threadIdx.x] - tsum;
#pragma unroll
  for (int k = 0; k < 4; ++k) {
    u32 i = base + threadIdx.x * 4u + k;
    if (i < n) out[i] = run;
    run += v[k];
  }
  if (threadIdx.x == TPB - 1 && partials) partials[blockIdx.x] = sh[TPB - 1];
}

__global__ void scan_small_kernel(u32* __restrict__ data, u32* __restrict__ total, u32 n) {
  __shared__ u32 sh[TPB];
  __shared__ u32 carry;
  if (threadIdx.x == 0) carry = 0;
  __syncthreads();
  for (u32 base = 0; base < n; base += 1024u) {
    u32 v[4]; u32 tsum = 0;
#pragma unroll
    for (int k = 0; k < 4; ++k) {
      u32 i = base + threadIdx.x * 4u + k;
      v[k] = (i < n) ? data[i] : 0u;
      tsum += v[k];
    }
    sh[threadIdx.x] = tsum;
    __syncthreads();
    for (u32 o = 1; o < TPB; o <<= 1) {
      u32 t = (threadIdx.x >= o) ? sh[threadIdx.x - o] : 0u;
      __syncthreads();
      sh[threadIdx.x] += t;
      __syncthreads();
    }
    u32 excl = sh[threadIdx.x] - tsum;
    u32 c = carry;
    __syncthreads();
    u32 run = c + excl;
#pragma unroll
    for (int k = 0; k < 4; ++k) {
      u32 i = base + threadIdx.x * 4u + k;
      if (i < n) data[i] = run;
      run += v[k];
    }
    if (threadIdx.x == TPB - 1) carry = c + sh[TPB - 1];
    __syncthreads();
  }
  if (total && threadIdx.x == 0) *total = carry;
}

__global__ void addback_kernel(u32* __restrict__ out, const u32* __restrict__ partials, u32 n) {
  u32 i = blockIdx.x * TPB + threadIdx.x;
  if (i < n) out[i] += partials[i >> 10];
}

// ------------------------------------------------------ unique / inv / decode
template <typename K>
__global__ void flag_heads_kernel(const K* __restrict__ skeys, u32* __restrict__ flags,
                                  u32 npad, K limit) {
  u32 i = blockIdx.x * TPB + threadIdx.x;
  if (i >= npad) return;
  K k = skeys[i];
  flags[i] = (k < limit && (i == 0 || skeys[i - 1] != k)) ? 1u : 0u;
}

__global__ void compute_inv_kernel(const u32* __restrict__ sidx, const u32* __restrict__ flags,
                                   const u32* __restrict__ ranks, u32* __restrict__ inv, u32 N) {
  u32 i = blockIdx.x * TPB + threadIdx.x;
  if (i >= N) return;
  inv[sidx[i]] = ranks[i] + flags[i] - 1u;  // rank of this element's unique key
}

__global__ void write_unique_nodes_kernel(const u32* __restrict__ skeys, const u32* __restrict__ flags,
                                          const u32* __restrict__ ranks, const u32* __restrict__ dmax,
                                          i64* __restrict__ uqpos, i64* __restrict__ nbatch, u32 N) {
  u32 i = blockIdx.x * TPB + threadIdx.x;
  if (i >= N || !flags[i]) return;
  u32 key = skeys[i], g = ranks[i];
  u32 kx = dmax[0] + 1u, ky = dmax[1] + 1u, kz = dmax[2] + 1u;
  u32 z = key % kz; u32 r = key / kz;
  u32 y = r % ky;   r /= ky;
  u32 qx = r % kx;  u32 b = r / kx;
  uqpos[3ull * g + 0] = (i64)qx;
  uqpos[3ull * g + 1] = (i64)y;
  uqpos[3ull * g + 2] = (i64)z;
  nbatch[g] = (i64)b;
}

__global__ void fill_invalid_nodes_kernel(i64* __restrict__ uqpos, i64* __restrict__ nbatch,
                                          const u32* __restrict__ counts, u32 N) {
  u32 g = blockIdx.x * TPB + threadIdx.x;
  if (g >= N) return;
  if (g >= counts[0]) {
    uqpos[3ull * g + 0] = -1; uqpos[3ull * g + 1] = -1; uqpos[3ull * g + 2] = -1;
    nbatch[g] = -1;
  }
}

__global__ void write_nuniq_kernel(const u32* __restrict__ counts, i64* __restrict__ out) {
  if (blockIdx.x == 0 && threadIdx.x == 0) out[0] = (i64)counts[0];
}

// --------------------------------------- segment max (TDM + async-LDS staging)
// pooled holds monotonic u32 encoding of f32: f>=0 -> bits|0x80000000, f<0 -> ~bits.
// mode==1 (N*F % 2048 == 0, F%4==0): one TENSOR_LOAD_TO_LDS per block stages the
// 8 KB x-tile (TENSORcnt), lanes 0..rows/4 async-stage the inv slice (ASYNCcnt).
// mode==0: generic per-lane async b128 staging with clamping.
__global__ void scatter_max_kernel(const float* __restrict__ x, const u32* __restrict__ inv,
                                   u32* __restrict__ pooled, u32 N, u32 F, u32 numTiles,
                                   u32 mode) {
  __shared__ float tile[2048];  // 8 KB staging tile
  __shared__ u32 invt[512];     // staged inv slice (mode 1; rows <= 512 since F>=4)
  const u32 rows = 2048u / F;   // rows per block (F=64 -> 32)
  const u32 total = rows * F;
  u64 base_row = (u64)blockIdx.x * rows;
  u32 ldsbase = (u32)(size_t)&tile[0];  // low 32 bits = LDS byte offset

  if (mode != 0u) {
    if (threadIdx.x < 32u) {  // wave 0 issues the TDM descriptor
      u64 gaddr = (u64)(size_t)x + (u64)blockIdx.x * 8192ull;  // tile byte address
      u32x4 g0;
      g0[0] = 1u;                      // count=1, is_restore=0, gather off
      g0[1] = ldsbase;                 // lds_addr
      g0[2] = (u32)gaddr;              // global_addr[31:0]
      g0[3] = (u32)((gaddr >> 32) & 0x01FFFFFFull) | 0x80000000u;  // addr[56:32] | type=2
      u32x8 g1;
      g1[0] = 0x00020000u;                              // data_size=4B; no mask/pad/iterate
      g1[1] = (2048u & 0xFFFFu) << 16;                  // tensor_dim0[15:0] = 2048
      g1[2] = (2048u >> 16) | ((numTiles & 0xFFFFu) << 16);  // dim0[31:16] | dim1[15:0]
      g1[3] = (numTiles >> 16) | (2048u << 16);         // dim1[31:16] | tile_dim0 = 2048
      g1[4] = 1u;                                       // tile_dim1=1, tile_dim2=0
      g1[5] = 2048u;                                    // tensor_dim0_stride[31:0]
      g1[6] = (2048u & 0xFFFFu) << 16;                  // stride0[47:32]=0 | stride1[15:0]
      g1[7] = 0u;                                       // stride1[47:16]
      asm volatile("tensor_load_to_lds %0, %1" : : "s"(g0), "s"(g1) : "memory");
    }
    u32 nchI = rows >> 2;  // 128-bit chunks of the inv slice
    if (threadIdx.x < nchI) {
      const u32* gi = inv + base_row + (u64)threadIdx.x * 4u;
      asm volatile("global_load_async_to_lds_b128 %0, %1, off"
                   :
                   : "v"((u32)(size_t)&invt[0] + threadIdx.x * 16u), "v"(gi)
                   : "memory");
    }
    wait_async0();
    wait_tensor0();
    __syncthreads();
    for (u32 t = threadIdx.x; t < total; t += TPB) {
      u64 r = base_row + (u64)(t / F);
      if (r < N) {  // always true in mode 1 (exact tiling); kept for safety
        u32 g = invt[t / F];
        u32 b = __float_as_uint(tile[t]);
        u32 m = (b & 0x80000000u) ? ~b : (b | 0x80000000u);
        atomicMax(&pooled[(u64)g * F + (t % F)], m);  // global_atomic_max_u32 in L2
      }
    }
  } else {
    const u32 nch = total / 4u;  // b128 chunks (4 floats each)
    for (u32 c = threadIdx.x; c < nch; c += TPB) {
      u64 r = base_row + (u64)((c * 4u) / F);
      if (r >= N) r = (u64)N - 1;  // clamp: safe duplicate read, atomics skipped below
      const float* g = x + r * (u64)F + (u64)((c * 4u) % F);
      asm volatile("global_load_async_to_lds_b128 %0, %1, off"
                   :
                   : "v"(ldsbase + c * 16u), "v"(g)
                   : "memory");
    }
    wait_async0();
    __syncthreads();
    for (u32 t = threadIdx.x; t < total; t += TPB) {
      u64 r = base_row + (u64)(t / F);
      if (r < N) {
        u32 g = inv[r];
        u32 b = __float_as_uint(tile[t]);
        u32 m = (b & 0x80000000u) ? ~b : (b | 0x80000000u);
        atomicMax(&pooled[(u64)g * F + (t % F)], m);
      }
    }
  }
}

__global__ void decode_pooled_kernel(u32* __restrict__ pooled, const u32* __restrict__ counts,
                                     u64 NF, u32 F) {
  u64 i = (u64)blockIdx.x * TPB + threadIdx.x;
  if (i >= NF) return;
  u32 g = (u32)(i / F);
  float v = 0.0f;
  if (g < counts[0]) {
    u32 m = pooled[i];
    u32 b = (m & 0x80000000u) ? (m & 0x7FFFFFFFu) : ~m;
    v = __uint_as_float(b);
  }
  ((float*)pooled)[i] = v;
}

// --------------------------------------------------------------- edge dedup
__global__ void make_ekeys_kernel(const i64* __restrict__ ei, const u32* __restrict__ inv,
                                  u64* __restrict__ keys, u32 E, u32 Epad, u64 n, u64 sentinel) {
  u32 i = blockIdx.x * TPB + threadIdx.x;
  if (i >= Epad) return;
  if (i < E) {
    u32 a = inv[(u64)ei[2ull * i + 0]];
    u32 b = inv[(u64)ei[2ull * i + 1]];
    keys[i] = (a != b) ? ((u64)a * n + (u64)b) : sentinel;
  } else {
    keys[i] = sentinel;
  }
}

__global__ void write_edges_kernel(const u64* __restrict__ skeys, const u32* __restrict__ flags,
                                   const u32* __restrict__ ranks, i64* __restrict__ nedge,
                                   u64 n, u32 Epad) {
  u32 i = blockIdx.x * TPB + threadIdx.x;
  if (i >= Epad || !flags[i]) return;
  u64 k = skeys[i];
  u32 g = ranks[i];
  nedge[2ull * g + 0] = (i64)(k / n);
  nedge[2ull * g + 1] = (i64)(k % n);
}

__global__ void fill_invalid_edges_kernel(i64* __restrict__ nedge, const u32* __restrict__ counts,
                                          u32 E) {
  u32 i = blockIdx.x * TPB + threadIdx.x;
  if (i >= E) return;
  if (i >= counts[1]) { nedge[2ull * i + 0] = -1; nedge[2ull * i + 1] = -1; }
}

__global__ void write_loops_kernel(i64* __restrict__ nedge, const u32* __restrict__ counts,
                                   u32 E, u32 N) {
  u32 j = blockIdx.x * TPB + threadIdx.x;
  if (j >= N) return;
  i64 v = (j < counts[0]) ? (i64)j : (i64)-1;
  u64 r = (u64)E + j;
  nedge[2ull * r + 0] = v;
  nedge[2ull * r + 1] = v;
}

// ---------------------------------------------------------------- launcher
extern "C" void kernel_launch(void* const* d_in, const int* in_sizes, int n_in,
                              void* d_out, int out_size, void* d_ws, size_t ws_size,
                              hipStream_t stream) {
  (void)n_in; (void)out_size; (void)ws_size;
  const float* x    = (const float*)d_in[0];
  const i64*  pos   = (const i64*)d_in[1];
  const i64*  ei    = (const i64*)d_in[2];
  const i64*  batch = (const i64*)d_in[3];

  const u32 N = (u32)in_sizes[3];
  const u32 F = (u32)(in_sizes[0] / in_sizes[3]);
  const u32 E = (u32)(in_sizes[2] / 2);
  const u32 Npad = (N + 255u) & ~255u;
  const u32 Epad = (E + 255u) & ~255u;
  const u64 sentinel = (u64)N * (u64)N;  // < 2^48 for N=1e6 -> 6 radix passes

  // Output layout (byte-contiguous, reference return order):
  // pooled f32[N*F] | uniq_qpos i64[N*3] | new_batch i64[N] | new_edge_index i64[(E+N)*2] | n_unique i64[1]
  char* ob = (char*)d_out;
  u32* pooled_u = (u32*)ob;
  i64* uqpos    = (i64*)(ob + (u64)N * F * 4u);
  i64* nbatch   = uqpos + (u64)N * 3u;
  i64* nedge    = nbatch + N;
  i64* nuq      = nedge + ((u64)E + N) * 2u;

  // Workspace carve (~84 MB for N=1e6, E=2e6)
  size_t off = 0;
  auto carve = [&](size_t bytes) -> char* {
    off = (off + 255u) & ~(size_t)255u;
    char* p = (char*)d_ws + off;
    off += bytes;
    return p;
  };
  u32* dmax     = (u32*)carve(16);
  u32* counts   = (u32*)carve(16);
  u32* partials = (u32*)carve(8192 * 4);
  u32* nkA = (u32*)carve((u64)Npad * 4);
  u32* nkB = (u32*)carve((u64)Npad * 4);
  u32* niA = (u32*)carve((u64)Npad * 4);
  u32* niB = (u32*)carve((u64)Npad * 4);
  u32* inv = (u32*)carve((u64)N * 4);
  u32* flagsN = (u32*)carve((u64)Npad * 4);
  u32* ranksN = (u32*)carve((u64)Npad * 4);
  u32* histN  = (u32*)carve((u64)Npad * 4);   // 256 * (Npad/256)
  u64* ekA = (u64*)carve((u64)Epad * 8);
  u64* ekB = (u64*)carve((u64)Epad * 8);
  u32* flagsE = (u32*)carve((u64)Epad * 4);
  u32* ranksE = (u32*)carve((u64)Epad * 4);
  u32* histE  = (u32*)carve((u64)Epad * 4);

  auto xscan = [&](const u32* in, u32* out, u32 n, u32* total) {
    u32 nb = (n + 1023u) / 1024u;
    scan_blocks_kernel<<<nb, TPB, 0, stream>>>(in, out, partials, n);
    scan_small_kernel<<<1, TPB, 0, stream>>>(partials, total, nb);
    addback_kernel<<<(n + TPB - 1) / TPB, TPB, 0, stream>>>(out, partials, n);
  };

  // ---- node unique pipeline
  fill_u32_kernel<<<1, TPB, 0, stream>>>(dmax, 0u, 3);
  qmax_kernel<<<(N + TPB - 1) / TPB, TPB, 0, stream>>>(pos, dmax, N);
  make_keys_kernel<<<Npad / TPB, TPB, 0, stream>>>(pos, batch, dmax, nkA, niA, N, Npad);

  {
    u32 nb = Npad / 256u;
    u32 *ki = nkA, *ko = nkB, *pi = niA, *po = niB;
    for (int pass = 0; pass < 4; ++pass) {  // even # of passes -> final in nkA/niA
      u32 shift = (u32)pass * 8u;
      radix_hist_kernel<u32><<<nb, TPB, 0, stream>>>(ki, histN, nb, shift);
      xscan(histN, histN, nb * 256u, nullptr);
      radix_scatter_kernel<u32, true><<<nb, TPB, 0, stream>>>(ki, pi, ko, po, histN, nb, shift);
      u32* t;
      t = ki; ki = ko; ko = t;
      t = pi; pi = po; po = t;
    }
  }

  flag_heads_kernel<u32><<<Npad / TPB, TPB, 0, stream>>>(nkA, flagsN, Npad, 0xFFFFFFFFu);
  xscan(flagsN, ranksN, Npad, &counts[0]);  // counts[0] = n_unique
  compute_inv_kernel<<<(N + TPB - 1) / TPB, TPB, 0, stream>>>(niA, flagsN, ranksN, inv, N);
  write_nuniq_kernel<<<1, 32, 0, stream>>>(counts, nuq);
  fill_invalid_nodes_kernel<<<(N + TPB - 1) / TPB, TPB, 0, stream>>>(uqpos, nbatch, counts, N);
  write_unique_nodes_kernel<<<(N + TPB - 1) / TPB, TPB, 0, stream>>>(nkA, flagsN, ranksN, dmax,
                                                                     uqpos, nbatch, N);

  // ---- segment max pooling (encoded u32 atomics, TDM/async-LDS staging)
  {
    u64 NF = (u64)N * F;
    fill_u32_kernel<<<(u32)((NF + TPB - 1) / TPB), TPB, 0, stream>>>(pooled_u, 0x007FFFFFu, NF);  // enc(-inf)
    u32 rows = 2048u / F;
    u32 mode = (F >= 4u && (F % 4u) == 0u && F <= 2048u && (NF % 2048ull) == 0ull) ? 1u : 0u;
    u32 numTiles = mode ? (u32)(NF / 2048ull) : 0u;
    u32 nb = mode ? numTiles : (N + rows - 1u) / rows;
    scatter_max_kernel<<<nb, TPB, 0, stream>>>(x, inv, pooled_u, N, F, numTiles, mode);
    decode_pooled_kernel<<<(u32)((NF + TPB - 1) / TPB), TPB, 0, stream>>>(pooled_u, counts, NF, F);
  }

  // ---- edge dedup pipeline
  make_ekeys_kernel<<<Epad / TPB, TPB, 0, stream>>>(ei, inv, ekA, E, Epad, (u64)N, sentinel);
  {
    u32 nb = Epad / 256u;
    u64 *ki = ekA, *ko = ekB;
    for (int pass = 0; pass < 6; ++pass) {  // 48 bits cover sentinel; even -> final in ekA
      u32 shift = (u32)pass * 8u;
      radix_hist_kernel<u64><<<nb, TPB, 0, stream>>>(ki, histE, nb, shift);
      xscan(histE, histE, nb * 256u, nullptr);
      radix_scatter_kernel<u64, false><<<nb, TPB, 0, stream>>>(ki, (const u32*)nullptr, ko,
                                                               (u32*)nullptr, histE, nb, shift);
      u64* t = ki; ki = ko; ko = t;
    }
  }
  flag_heads_kernel<u64><<<Epad / TPB, TPB, 0, stream>>>(ekA, flagsE, Epad, sentinel);
  xscan(flagsE, ranksE, Epad, &counts[1]);  // counts[1] = #distinct non-self-loop edges
  write_edges_kernel<<<Epad / TPB, TPB, 0, stream>>>(ekA, flagsE, ranksE, nedge, (u64)N, Epad);
  fill_invalid_edges_kernel<<<(E + TPB - 1) / TPB, TPB, 0, stream>>>(nedge, counts, E);
  write_loops_kernel<<<(N + TPB - 1) / TPB, TPB, 0, stream>>>(nedge, counts, E, N);
}